// Attention_89163521065188
// MI455X (gfx1250) — compile-verified
//
#include <hip/hip_runtime.h>

#define D 128
#define SEGS 16384
#define K2_ROWS 128   // rows per block in logits kernel: 8 waves x 16 rows
#define K4_ROWS 128   // rows per block in scatter kernel

typedef __attribute__((ext_vector_type(2))) float v2f;
typedef __attribute__((ext_vector_type(8))) float v8f;

// Order-preserving float<->uint encoding so atomicMax(u32) == float max.
// Encoded 0 acts as -infinity (below every encoded real float).
__device__ __forceinline__ unsigned int enc_f32(float f) {
    unsigned int b = __float_as_uint(f);
    return (b & 0x80000000u) ? ~b : (b | 0x80000000u);
}
__device__ __forceinline__ float dec_f32(unsigned int u) {
    unsigned int b = (u & 0x80000000u) ? (u & 0x7fffffffu) : ~u;
    return __uint_as_float(b);
}

// ---------------------------------------------------------------------------
// Kernel 1: zero the output accumulator + per-segment buffers (every call,
// since d_out is accumulated into and the harness replays the graph).
// ---------------------------------------------------------------------------
__global__ void init_kernel(float* __restrict__ out,
                            float* __restrict__ segsum,
                            unsigned int* __restrict__ segmax,
                            int out_elems) {
    int tid = blockIdx.x * blockDim.x + threadIdx.x;
    if (tid < out_elems) out[tid] = 0.0f;
    if (tid < SEGS) { segsum[tid] = 0.0f; segmax[tid] = 0u; }
}

// ---------------------------------------------------------------------------
// Kernel 2: s_i = tanh([x_i; ref_i] . W + b)  via V_WMMA_F32_16X16X4_F32.
// One wave computes 16 rows; K=256 covered in 64 WMMA steps of K=4.
// A layout (32-bit 16x4): lane l -> row (l&15), holds K = kb+2*(l>>4) +{0,1}.
// B: W chunk replicated into all 16 columns, so every output column of the
// 16x16 accumulator carries the same dot product (we read column 0).
//
// Out-of-range rows: A row m only affects D row m, so we clamp the row
// pointer to a valid row (no divergent loads in the loop) and discard the
// garbage rows at store time with the rr < N check.
//
// Two accumulators (x-half / ref-half) interleave independent WMMAs so the
// matrix pipe is not serialized on one RAW accumulation chain.
// ---------------------------------------------------------------------------
__global__ void logits_kernel(const float* __restrict__ x,
                              const float* __restrict__ ref,
                              const float* __restrict__ W,
                              const float* __restrict__ bptr,
                              const int* __restrict__ index,
                              float* __restrict__ s_out,
                              unsigned int* __restrict__ segmax,
                              int N) {
    const int lane = threadIdx.x & 31;
    const int wave = threadIdx.x >> 5;
    const int mrow = lane & 15;
    const int kp   = lane >> 4;           // 0: K pair {0,1}, 1: K pair {2,3}
    const long r0  = (long)blockIdx.x * K2_ROWS + (long)wave * 16;
    const long row = r0 + mrow;
    // Clamp to a valid row: garbage results are discarded at store time.
    const long rowc = (row < (long)N) ? row : (long)(N - 1);

    const float* xrow = x   + rowc * D + kp * 2;
    const float* rrow = ref + rowc * D + kp * 2;
    const float* wrow = W   + kp * 2;

    v8f c0 = {0.f, 0.f, 0.f, 0.f, 0.f, 0.f, 0.f, 0.f};
    v8f c1 = {0.f, 0.f, 0.f, 0.f, 0.f, 0.f, 0.f, 0.f};

#pragma unroll 8
    for (int cc = 0; cc < 32; ++cc) {
        const int col = cc * 4;
        const float2 ax = *(const float2*)(xrow + col);        // x[row, col+2kp ..]
        const float2 ar = *(const float2*)(rrow + col);        // ref[row, col+2kp ..]
        const float2 bx = *(const float2*)(wrow + col);        // W[col+2kp ..]
        const float2 br = *(const float2*)(wrow + 128 + col);  // W[128+col+2kp ..]
        v2f a0; a0.x = ax.x; a0.y = ax.y;
        v2f b0; b0.x = bx.x; b0.y = bx.y;
        v2f a1; a1.x = ar.x; a1.y = ar.y;
        v2f b1; b1.x = br.x; b1.y = br.y;
        c0 = __builtin_amdgcn_wmma_f32_16x16x4_f32(false, a0, false, b0,
                                                   (short)0, c0, false, false);
        c1 = __builtin_amdgcn_wmma_f32_16x16x4_f32(false, a1, false, b1,
                                                   (short)0, c1, false, false);
    }

    // Extract column 0 of the 16x16 accumulator:
    // lane 0 holds rows 0..7 in c[0..7]; lane 16 holds rows 8..15.
    if ((lane & 15) == 0) {
        const float bias = bptr[0];
#pragma unroll
        for (int v = 0; v < 8; ++v) {
            const long rr = r0 + v + 8 * kp;
            if (rr < (long)N) {
                const float s = tanhf(c0[v] + c1[v] + bias);
                s_out[rr] = s;
                atomicMax(&segmax[index[rr]], enc_f32(s));
            }
        }
    }
}

// ---------------------------------------------------------------------------
// Kernel 3: e_i = exp(s_i - segmax[index_i]); segsum[index_i] += e_i.
// Overwrites s buffer in place with e.
// ---------------------------------------------------------------------------
__global__ void softmax_kernel(const int* __restrict__ index,
                               const unsigned int* __restrict__ segmax,
                               float* __restrict__ s_e,
                               float* __restrict__ segsum,
                               int N) {
    const long i = (long)blockIdx.x * blockDim.x + threadIdx.x;
    if (i >= (long)N) return;
    const int idx = index[i];
    const float m = dec_f32(segmax[idx]);
    const float e = expf(s_e[i] - m);
    s_e[i] = e;
    atomicAdd(&segsum[idx], e);
}

// ---------------------------------------------------------------------------
// Kernel 4: out[seg, :] += (e_i / (segsum[seg] + 1e-16)) * x[i, :].
// index is sorted, so accumulate runs of equal segment in registers and
// flush with one atomicAdd per (block, segment, column).
// 128 threads = 128 columns; each block walks 128 consecutive rows.
// ---------------------------------------------------------------------------
__global__ void scatter_kernel(const float* __restrict__ x,
                               const int* __restrict__ index,
                               const float* __restrict__ e,
                               const float* __restrict__ segsum,
                               float* __restrict__ out,
                               int N) {
    const int d = threadIdx.x;                       // column 0..127
    const long base = (long)blockIdx.x * K4_ROWS;
    float acc = 0.0f;
    int cur = -1;
    for (int r = 0; r < K4_ROWS; ++r) {
        const long i = base + r;
        if (i >= (long)N) break;                     // uniform across block
        const int idx = index[i];                    // uniform broadcast load
        const float w = e[i] / (segsum[idx] + 1e-16f);
        if (idx != cur) {                            // uniform branch
            if (cur >= 0) atomicAdd(&out[(long)cur * D + d], acc);
            cur = idx;
            acc = 0.0f;
        }
        acc = fmaf(w, x[i * D + d], acc);            // coalesced 512B/row
    }
    if (cur >= 0) atomicAdd(&out[(long)cur * D + d], acc);
}

// ---------------------------------------------------------------------------
// Host launcher. Inputs (setup_inputs order):
//   d_in[0]=x [N*128 f32], d_in[1]=ref [N*128 f32], d_in[2]=index [N i32],
//   d_in[3]=batch_size [1 i32], d_in[4]=W [256 f32], d_in[5]=b [1 f32].
// Output: [16384, 128] f32. Workspace: N f32 (s/e) + SEGS u32 + SEGS f32.
// ---------------------------------------------------------------------------
extern "C" void kernel_launch(void* const* d_in, const int* in_sizes, int n_in,
                              void* d_out, int out_size, void* d_ws, size_t ws_size,
                              hipStream_t stream) {
    const float* x     = (const float*)d_in[0];
    const float* ref   = (const float*)d_in[1];
    const int*   index = (const int*)d_in[2];
    const float* W     = (const float*)d_in[4];
    const float* b     = (const float*)d_in[5];
    const int N = in_sizes[2];

    float* out = (float*)d_out;

    float*        ws_s   = (float*)d_ws;                    // N floats (s, then e)
    unsigned int* segmax = (unsigned int*)(ws_s + N);       // SEGS u32
    float*        segsum = (float*)(segmax + SEGS);         // SEGS floats

    init_kernel<<<(out_size + 255) / 256, 256, 0, stream>>>(out, segsum, segmax, out_size);

    const int blocks2 = (N + K2_ROWS - 1) / K2_ROWS;
    logits_kernel<<<blocks2, 256, 0, stream>>>(x, ref, W, b, index, ws_s, segmax, N);

    softmax_kernel<<<(N + 255) / 256, 256, 0, stream>>>(index, segmax, ws_s, segsum, N);

    const int blocks4 = (N + K4_ROWS - 1) / K4_ROWS;
    scatter_kernel<<<blocks4, D, 0, stream>>>(x, index, ws_s, segsum, out, N);
}